// HashMapNGP_26130581029487
// MI455X (gfx1250) — compile-verified
//
#include <hip/hip_runtime.h>
#include <stdint.h>

// Instant-NGP 2D hash-grid encode, MI455X-tuned.
//
// Roofline: HBM traffic = 4MB coords + 64MB out + 64MB tables ≈ 132MB -> ~6us
// floor at 23.3 TB/s. All 16 hash tables (64MB) fit in the 192MB L2, so the
// 33.5M random b64 gathers are L2 hits — provided the 64MB write-once output
// stream doesn't evict them; output stores are therefore non-temporal.
// No matmul anywhere -> WMMA inapplicable; the CDNA5 features that fit are the
// async global->LDS copy (ASYNCcnt path) for coord staging and NT store hints.
//
// Hash: h = (x*1) ^ (y*2654435761) mod 2^19 depends only on low 19 bits, and
// low 32 bits of the int64 product == uint32 wraparound product -> pure u32 math.

namespace {
constexpr int      kLevels = 16;
constexpr unsigned kTMask  = (1u << 19) - 1u;   // T = 2^19
constexpr unsigned kPi2    = 2654435761u;
constexpr int      kBlock  = 256;
typedef float f32x4 __attribute__((ext_vector_type(4)));
}

__global__ __launch_bounds__(kBlock)
void ngp_hashgrid_kernel(const int* __restrict__ coords,
                         const float* __restrict__ tables,
                         float* __restrict__ out, int n)
{
    __shared__ int sc[2 * kBlock];

    const int tid = threadIdx.x;
    const int gid = blockIdx.x * kBlock + tid;
    const int cg  = (gid < n) ? gid : (n - 1);   // clamp for tail safety

    // ---- CDNA5 async copy: stage this block's coords into LDS -------------
    // GVS mode: each lane copies 8 bytes from global saddr plus its vaddr offset to the LDS address in vdst. Tracked by ASYNCcnt;
    // keeps the VGPR load-return path free for the table gathers.
    {
        unsigned ldsOff = (unsigned)(uintptr_t)(sc + 2 * tid); // lds byte addr
        unsigned gOff   = (unsigned)cg * 8u;                   // byte offset
        asm volatile("global_load_async_to_lds_b64 %0, %1, %2"
                     :: "v"(ldsOff), "v"(gOff), "s"(coords)
                     : "memory");
        asm volatile("s_wait_asynccnt 0" ::: "memory");
    }

    const float cx = (float)sc[2 * tid + 0];
    const float cy = (float)sc[2 * tid + 1];

    // scale[l] = 16 * b^l / 1024 = 2^((2l-30)/5),  b = exp((ln1024-ln16)/15)
    const float kScale[kLevels] = {
        0.015625f,     0.0206173111f, 0.0272047051f, 0.0358968236f,
        0.0473661427f, 0.0625f,       0.0824692444f, 0.1088188204f,
        0.1435872944f, 0.1894645708f, 0.25f,         0.3298769777f,
        0.4352752816f, 0.5743491775f, 0.7578582833f, 1.0f
    };

    float r[2 * kLevels];

#pragma unroll
    for (int l = 0; l < kLevels; ++l) {
        const float s = kScale[l];

        float sx, sy, rw, cw;
        int   x0, y0, x1, y1;
        {
            // Index-critical math: must match the reference's rounding
            // sequence exactly (round cx*s, THEN add 1, THEN trunc).
            // Forbid FMA contraction here — a fused cx*s+1 differs by 1 ULP
            // precisely at the cell-boundary points where the corner index
            // (and thus the gathered hash entry) changes.
#pragma clang fp contract(off)
            sx = cx * s;
            sy = cy * s;
            x0 = (int)sx;                // trunc(scaled + 0)
            y0 = (int)sy;
            x1 = (int)(sx + 1.0f);       // trunc(scaled + 1), fp add first
            y1 = (int)(sy + 1.0f);
            rw = sx - floorf(sx);        // frac along dim0
            cw = sy - floorf(sy);        // frac along dim1
        }

        const unsigned hy0 = (unsigned)y0 * kPi2;
        const unsigned hy1 = (unsigned)y1 * kPi2;
        const unsigned i00 = ((unsigned)x0 ^ hy0) & kTMask;  // off (0,0)
        const unsigned i01 = ((unsigned)x0 ^ hy1) & kTMask;  // off (0,1)
        const unsigned i11 = ((unsigned)x1 ^ hy1) & kTMask;  // off (1,1)
        const unsigned i10 = ((unsigned)x1 ^ hy0) & kTMask;  // off (1,0)

        const float2* __restrict__ tab =
            (const float2*)tables + (size_t)l * (size_t)(kTMask + 1u);
        const float2 v00 = tab[i00];   // L2-resident b64 gathers (default TH)
        const float2 v01 = tab[i01];
        const float2 v11 = tab[i11];
        const float2 v10 = tab[i10];

        const float w00 = (1.0f - cw) * (1.0f - rw);
        const float w01 = cw * (1.0f - rw);
        const float w10 = (1.0f - cw) * rw;
        const float w11 = cw * rw;

        r[2 * l + 0] = v00.x * w00 + v01.x * w01 + v10.x * w10 + v11.x * w11;
        r[2 * l + 1] = v00.y * w00 + v01.y * w01 + v10.y * w10 + v11.y * w11;
    }

    // Each thread owns a contiguous 128B output row -> 8x b128 stores.
    // Non-temporal: output is write-once, keep it from evicting the
    // L2-resident hash tables.
    if (gid < n) {
        f32x4* __restrict__ o = (f32x4*)out + (size_t)gid * 8;
#pragma unroll
        for (int j = 0; j < 8; ++j) {
            f32x4 v = { r[4 * j + 0], r[4 * j + 1], r[4 * j + 2], r[4 * j + 3] };
            __builtin_nontemporal_store(v, &o[j]);
        }
    }
}

extern "C" void kernel_launch(void* const* d_in, const int* in_sizes, int n_in,
                              void* d_out, int out_size, void* d_ws, size_t ws_size,
                              hipStream_t stream) {
    const int*   coords = (const int*)d_in[0];     // (N, 2) int32
    const float* tables = (const float*)d_in[1];   // (16, 2^19, 2) f32
    float*       out    = (float*)d_out;           // (N, 32) f32

    const int n      = in_sizes[0] / 2;            // 524288 coords
    const int blocks = (n + kBlock - 1) / kBlock;
    ngp_hashgrid_kernel<<<blocks, kBlock, 0, stream>>>(coords, tables, out, n);
}